// DynamicCore_33320356282810
// MI455X (gfx1250) — compile-verified
//
#include <hip/hip_runtime.h>

typedef __attribute__((ext_vector_type(2))) float v2f;
typedef __attribute__((ext_vector_type(8))) float v8f;

namespace {
constexpr int BDIM = 2, YDIM = 360, XDIM = 720, ZDIM = 32;
constexpr int NCOL = BDIM * YDIM * XDIM;   // 518400
constexpr int N4D  = NCOL * ZDIM;          // 16588800
constexpr int NTILE = XDIM / 16;           // 45 output tiles per row-group
constexpr float RADe = 6371000.0f;
constexpr float OMG  = 7.292e-5f;
constexpr float CPc  = 1004.0f;
constexpr float PIf  = 3.14159265358979323846f;
constexpr float DLAT = PIf / YDIM;
constexpr float DLON = 2.0f * PIf / XDIM;
constexpr float DS   = 1.0f / ZDIM;
}

// ---------------- wave32 scan helpers (lane == z) ----------------
__device__ __forceinline__ float wscan_inc(float v, int lane) {
#pragma unroll
  for (int off = 1; off < 32; off <<= 1) {
    float n = __shfl_up(v, off, 32);
    if (lane >= off) v += n;
  }
  return v;
}
__device__ __forceinline__ float suffix_sum(float g, int lane) {
  float inc = wscan_inc(g, lane);
  float tot = __shfl(inc, 31, 32);
  return tot - inc + g;   // inclusive suffix sum (cumsum from bottom)
}
__device__ __forceinline__ float vgrad32(float f, int z) {
  float fup = __shfl_down(f, 1, 32);
  float fdn = __shfl_up(f, 1, 32);
  if (z == 0)  return (fup - f) * (1.0f / DS);
  if (z == 31) return (f - fdn) * (1.0f / DS);
  return (fup - fdn) * (0.5f / DS);
}

// ---------------- Dirichlet kernel table: h[y][d] ----------------
__global__ void build_htab_kernel(float* __restrict__ htab) {
  int i = blockIdx.x * blockDim.x + threadIdx.x;
  if (i >= YDIM * XDIM) return;
  int y = i / XDIM, d = i - y * XDIM;
  float lat = -0.5f * PIf + ((float)y + 0.5f) * DLAT;
  float cut = fmaxf(1.0f, (float)(XDIM / 2) * cosf(lat));
  int K = (int)floorf(cut);
  if (K > XDIM / 2) K = XDIM / 2;
  float h;
  if (d == 0) {
    h = (float)(2 * K + 1) / (float)XDIM;
  } else {
    // exact integer phase reduction: sin((K+1/2)*2*pi*d/X) with period 2X
    int mnum = (d * (2 * K + 1)) % (2 * XDIM);
    float num = sinf(PIf * (float)mnum / (float)XDIM);
    float den = sinf(PIf * (float)d / (float)XDIM);
    h = num / ((float)XDIM * den);
  }
  htab[i] = h;
}

// ---------------- ps horizontal gradients ----------------
__global__ void psgrad_kernel(const float* __restrict__ ps,
                              float* __restrict__ dpsdx,
                              float* __restrict__ dpsdy) {
  int i = blockIdx.x * blockDim.x + threadIdx.x;
  if (i >= NCOL) return;
  int x = i % XDIM;
  int t = i / XDIM;          // t = b*YDIM + y
  int y = t % YDIM;
  float lat = -0.5f * PIf + ((float)y + 0.5f) * DLAT;
  float cl = cosf(lat);
  int xm = (x == 0) ? XDIM - 1 : x - 1;
  int xp = (x == XDIM - 1) ? 0 : x + 1;
  int tb = t * XDIM;
  float pc  = ps[i];
  float pxm = ps[tb + xm], pxp = ps[tb + xp];
  float pym = (y > 0) ? ps[i - XDIM] : pc;
  float pyp = (y < YDIM - 1) ? ps[i + XDIM] : pc;
  float inv2dx = 1.0f / (2.0f * DLON * RADe * cl);
  float invdy1 = 1.0f / (DLAT * RADe);
  dpsdx[i] = (pxp - pxm) * inv2dx;
  float gy;
  if (y == 0)             gy = (pyp - pc) * invdy1;
  else if (y == YDIM - 1) gy = (pc - pym) * invdy1;
  else                    gy = (pyp - pym) * (0.5f * invdy1);
  dpsdy[i] = gy;
}

// ---------------- pass 1: div_h = du/dx + dv/dy ----------------
__global__ __launch_bounds__(256) void divh_kernel(const float* __restrict__ u,
                                                   const float* __restrict__ v,
                                                   float* __restrict__ divh) {
  int c = blockIdx.x * 8 + (threadIdx.x >> 5);
  int z = threadIdx.x & 31;
  int x = c % XDIM;
  int t = c / XDIM;
  int y = t % YDIM;
  float lat = -0.5f * PIf + ((float)y + 0.5f) * DLAT;
  float cl = cosf(lat);
  int xm = (x == 0) ? XDIM - 1 : x - 1;
  int xp = (x == XDIM - 1) ? 0 : x + 1;
  int tym = (y > 0) ? t - 1 : t;
  int typ = (y < YDIM - 1) ? t + 1 : t;
  int tb = t * XDIM;
  int ec  = (tb + x) * ZDIM + z;
  int exm = (tb + xm) * ZDIM + z;
  int exp_ = (tb + xp) * ZDIM + z;
  int eym = (tym * XDIM + x) * ZDIM + z;
  int eyp = (typ * XDIM + x) * ZDIM + z;
  float inv2dx = 1.0f / (2.0f * DLON * RADe * cl);
  float invdy1 = 1.0f / (DLAT * RADe);
  float dudx = (u[exp_] - u[exm]) * inv2dx;
  float vc = v[ec], vym = v[eym], vyp = v[eyp];
  float dvdy;
  if (y == 0)             dvdy = (vyp - vc) * invdy1;
  else if (y == YDIM - 1) dvdy = (vc - vym) * invdy1;
  else                    dvdy = (vyp - vym) * (0.5f * invdy1);
  divh[ec] = dudx + dvdy;
}

// ---------------- pass 2: raw tendencies (pre-filter) ----------------
__global__ __launch_bounds__(256) void dyn_core_kernel(
    const float* __restrict__ u, const float* __restrict__ v,
    const float* __restrict__ T, const float* __restrict__ ps,
    const float* __restrict__ divh,
    const float* __restrict__ dpsdx_g, const float* __restrict__ dpsdy_g,
    const float* __restrict__ Rptr, float* __restrict__ out) {
  int c = blockIdx.x * 8 + (threadIdx.x >> 5);
  int z = threadIdx.x & 31;
  float Rc = Rptr[0];
  int x = c % XDIM;
  int t = c / XDIM;          // b*YDIM + y
  int y = t % YDIM;
  float lat = -0.5f * PIf + ((float)y + 0.5f) * DLAT;
  float cl = cosf(lat);
  int xm = (x == 0) ? XDIM - 1 : x - 1;
  int xp = (x == XDIM - 1) ? 0 : x + 1;
  int tym = (y > 0) ? t - 1 : t;
  int typ = (y < YDIM - 1) ? t + 1 : t;
  int tb = t * XDIM;
  int ec  = (tb + x) * ZDIM + z;
  int exm = (tb + xm) * ZDIM + z;
  int exp_ = (tb + xp) * ZDIM + z;
  int eym = (tym * XDIM + x) * ZDIM + z;
  int eyp = (typ * XDIM + x) * ZDIM + z;

  float uc = u[ec],  uxm = u[exm],  uxp = u[exp_],  uym = u[eym],  uyp = u[eyp];
  float vc = v[ec],  vxm = v[exm],  vxp = v[exp_],  vym = v[eym],  vyp = v[eyp];
  float Tc = T[ec],  Txm = T[exm],  Txp = T[exp_],  Tym = T[eym],  Typ = T[eyp];
  float dc = divh[ec], dxm = divh[exm], dxp = divh[exp_], dym = divh[eym], dyp = divh[eyp];

  // hydrostatic geopotential at the 5 stencil columns (suffix cumsum over z)
  float invzp = 1.0f / ((float)z + 0.5f);     // ds/b_k = 1/(z+0.5)
  float phc  = suffix_sum(Rc * Tc  * invzp, z);
  float phxm = suffix_sum(Rc * Txm * invzp, z);
  float phxp = suffix_sum(Rc * Txp * invzp, z);
  float phym = suffix_sum(Rc * Tym * invzp, z);
  float phyp = suffix_sum(Rc * Typ * invzp, z);

  float inv2dx = 1.0f / (2.0f * DLON * RADe * cl);
  float invdy1 = 1.0f / (DLAT * RADe);
  float invdy2 = 0.5f * invdy1;
  float cxi = 1.0f / (DLON * RADe * cl);
  float cx2 = cxi * cxi;
  float cyi = 1.0f / (DLAT * RADe);
  float cy2 = cyi * cyi;
  bool edge = (y == 0) || (y == YDIM - 1);

  auto dfy = [&](float fc, float fm, float fp) -> float {
    if (y == 0)        return (fp - fc) * invdy1;
    if (y == YDIM - 1) return (fc - fm) * invdy1;
    return (fp - fm) * invdy2;
  };
  auto lap = [&](float fc, float fxm_, float fxp_, float fym_, float fyp_) -> float {
    float d2x = (fxp_ + fxm_ - 2.0f * fc) * cx2;
    float d2y = edge ? 0.0f : (fyp_ + fym_ - 2.0f * fc) * cy2;
    return d2x + d2y;
  };

  float dudx = (uxp - uxm) * inv2dx,   dudy = dfy(uc, uym, uyp);
  float dvdx = (vxp - vxm) * inv2dx,   dvdy = dfy(vc, vym, vyp);
  float dTdx = (Txp - Txm) * inv2dx,   dTdy = dfy(Tc, Tym, Typ);
  float dphidx = (phxp - phxm) * inv2dx, dphidy = dfy(phc, phym, phyp);
  float dgx = (dxp - dxm) * inv2dx,    dgy = dfy(dc, dym, dyp);

  float psc = ps[tb + x];
  float gxp = dpsdx_g[c], gyp_ = dpsdy_g[c];

  // mass column, dpsdt, sigma_dot
  float mass = dc * psc + uc * gxp + vc * gyp_;
  float Iinc = wscan_inc(mass * DS, z);
  float I31  = __shfl(Iinc, 31, 32);
  float bk = ((float)z + 0.5f) / (float)ZDIM;
  float sd = (bk * I31 - Iinc) / (psc + 1e-8f);
  if (z == 0) out[4 * N4D + c] = -I31;   // raw dpsdt

  float p = bk * psc;
  float fcor = 2.0f * OMG * sinf(lat);
  float cf = fmaxf(cl * cl, 0.01f);
  float nu_div = 500000.0f * cf;
  float nu_h   = 100000.0f * cf;

  float pgf_u = -dphidx - Rc * Tc * bk / (p + 1e-8f) * gxp;
  float pgf_v = -dphidy - Rc * Tc * bk / (p + 1e-8f) * gyp_;
  float adiab = Rc * Tc / (p * CPc + 1e-8f) * (sd * psc);

  float vgu = vgrad32(uc, z), vgv = vgrad32(vc, z), vgT = vgrad32(Tc, z);

  float dudt = -(uc * dudx + vc * dudy) - sd * vgu + fcor * vc + pgf_u
             + nu_h * lap(uc, uxm, uxp, uym, uyp) - 1e-5f * uc + nu_div * dgx;
  float dvdt = -(uc * dvdx + vc * dvdy) - sd * vgv - fcor * uc + pgf_v
             + nu_h * lap(vc, vxm, vxp, vym, vyp) - 1e-5f * vc + nu_div * dgy;
  float dTdt = -(uc * dTdx + vc * dTdy) - sd * vgT + adiab
             + nu_h * lap(Tc, Txm, Txp, Tym, Typ);

  out[ec]            = dudt;
  out[N4D + ec]      = dvdt;
  out[2 * N4D + ec]  = dTdt;
}

// ---------------- dqdt = 0 ----------------
__global__ void zero4_kernel(float4* __restrict__ p, int n) {
  int i = blockIdx.x * blockDim.x + threadIdx.x;
  if (i < n) p[i] = make_float4(0.f, 0.f, 0.f, 0.f);
}

// ---------------- pass 3: polar filter via circulant WMMA matmul ----------------
// One block per (field, y, rowgroup). Rowgroup = 16 rows (b fixed, z0..z0+15),
// each row = 720 x-values strided by ZDIM floats. OUT = IN @ M_y, M_y circulant
// from Dirichlet kernel h_y. In-place on d_out (input fully staged in LDS).
// Wave w owns tiles {w, w+8, ...}: tiles 0..4 are unconditional (w+32 < 45),
// only the 6th is guarded — and the guard is SCALAR (readfirstlane'd wave id),
// so no EXEC manipulation appears in the WMMA loop.
__global__ __launch_bounds__(256) void polar_filter4d_kernel(float* __restrict__ out,
                                                             const float* __restrict__ htab) {
  __shared__ float Alds[XDIM * 16];    // [x][row]  46080 B
  __shared__ float g2[2 * XDIM];       // reversed doubled Dirichlet kernel
  int bid = blockIdx.x;
  int rg = bid & 3;
  int rem = bid >> 2;
  int y = rem % YDIM;
  int field = rem / YDIM;              // 0:u 1:v 2:T
  float lim = (field == 2) ? 0.01f : 0.02f;
  float* f = out + (size_t)field * N4D;
  int b = rg >> 1, z0 = (rg & 1) * 16;
  size_t colbase = (size_t)(b * YDIM + y) * XDIM * ZDIM;
  int tid = threadIdx.x;

  // g2[j] = h[(2X-1-j) mod X]  =>  B[k][n] = h2[hi] = g2[2X-1-hi]
  for (int j = tid; j < 2 * XDIM; j += 256) {
    int src = (2 * XDIM - 1 - j) % XDIM;
    g2[j] = htab[y * XDIM + src];
  }
  int r = tid & 15, xq = tid >> 4;
  for (int it = 0; it < NTILE; ++it) {           // coalesced: 16 consecutive z
    int xx = it * 16 + xq;
    Alds[xx * 16 + r] = f[colbase + (size_t)xx * ZDIM + z0 + r];
  }
  __syncthreads();

  int wave = __builtin_amdgcn_readfirstlane(tid >> 5);  // SGPR: scalar control flow
  int lane = tid & 31;
  int m = lane & 15, kh = lane >> 4;             // A: M=m ; B: N=m ; K-half = kh
  bool has6 = (wave < NTILE - 40);               // scalar, loop-invariant

  v8f acc[6];
#pragma unroll
  for (int i = 0; i < 6; ++i) acc[i] = (v8f){0.f,0.f,0.f,0.f,0.f,0.f,0.f,0.f};

  for (int kg = 0; kg < XDIM; kg += 4) {
    int ka = kg + kh * 2;
    v2f a;                                       // shared across this wave's tiles
    a.x = Alds[ka * 16 + m];
    a.y = Alds[(ka + 1) * 16 + m];
    int bb = (XDIM - 1) - m + ka;                // g2 base: 719 - x0 - m + ka
#pragma unroll
    for (int i = 0; i < 5; ++i) {                // always valid: wave+8*4 <= 39 < 45
      int idx = bb - (wave + 8 * i) * 16;
      v2f bm; bm.x = g2[idx]; bm.y = g2[idx + 1];
      acc[i] = __builtin_amdgcn_wmma_f32_16x16x4_f32(false, a, false, bm,
                                                     (short)0, acc[i], false, false);
    }
    if (has6) {                                  // scalar branch: EXEC untouched
      int idx = bb - (wave + 40) * 16;
      v2f bm; bm.x = g2[idx]; bm.y = g2[idx + 1];
      acc[5] = __builtin_amdgcn_wmma_f32_16x16x4_f32(false, a, false, bm,
                                                     (short)0, acc[5], false, false);
    }
  }

  int mb = kh * 8;                               // D rows m_base..m_base+7 -> contiguous z
#pragma unroll
  for (int i = 0; i < 6; ++i) {
    if (i < 5 || has6) {
      int tile = wave + 8 * i;
      size_t addr = colbase + (size_t)(tile * 16 + m) * ZDIM + z0 + mb;
      float4 lo = make_float4(fminf(fmaxf(acc[i][0], -lim), lim),
                              fminf(fmaxf(acc[i][1], -lim), lim),
                              fminf(fmaxf(acc[i][2], -lim), lim),
                              fminf(fmaxf(acc[i][3], -lim), lim));
      float4 hi4 = make_float4(fminf(fmaxf(acc[i][4], -lim), lim),
                               fminf(fmaxf(acc[i][5], -lim), lim),
                               fminf(fmaxf(acc[i][6], -lim), lim),
                               fminf(fmaxf(acc[i][7], -lim), lim));
      *(float4*)(f + addr)     = lo;
      *(float4*)(f + addr + 4) = hi4;
    }
  }
}

// ---------------- polar filter for dpsdt (2*360 rows, scalar conv) ----------------
__global__ __launch_bounds__(256) void polar_filter_ps_kernel(float* __restrict__ dps,
                                                              const float* __restrict__ htab) {
  __shared__ float row[XDIM];
  __shared__ float h2[2 * XDIM];
  int y = blockIdx.x % YDIM;
  int b = blockIdx.x / YDIM;
  float* rp = dps + (size_t)(b * YDIM + y) * XDIM;
  int tid = threadIdx.x;
  for (int j = tid; j < XDIM; j += 256) {
    row[j] = rp[j];
    float hv = htab[y * XDIM + j];
    h2[j] = hv; h2[j + XDIM] = hv;
  }
  __syncthreads();
  for (int xx = tid; xx < XDIM; xx += 256) {
    float s = 0.f;
#pragma unroll 4
    for (int xs = 0; xs < XDIM; ++xs) s += row[xs] * h2[xx - xs + XDIM];
    rp[xx] = fminf(fmaxf(s, -0.5f), 0.5f);
  }
}

extern "C" void kernel_launch(void* const* d_in, const int* in_sizes, int n_in,
                              void* d_out, int out_size, void* d_ws, size_t ws_size,
                              hipStream_t stream) {
  (void)in_sizes; (void)n_in; (void)out_size; (void)ws_size;
  const float* u  = (const float*)d_in[0];
  const float* v  = (const float*)d_in[1];
  const float* T  = (const float*)d_in[2];
  const float* ps = (const float*)d_in[4];
  const float* Rp = (const float*)d_in[6];
  float* out = (float*)d_out;
  float* ws  = (float*)d_ws;

  float* divh  = ws;                         // N4D
  float* dpsdx = ws + (size_t)N4D;           // NCOL
  float* dpsdy = ws + (size_t)N4D + NCOL;    // NCOL
  float* htab  = ws + (size_t)N4D + 2 * NCOL;// YDIM*XDIM

  build_htab_kernel<<<(YDIM * XDIM + 255) / 256, 256, 0, stream>>>(htab);
  psgrad_kernel<<<(NCOL + 255) / 256, 256, 0, stream>>>(ps, dpsdx, dpsdy);
  divh_kernel<<<NCOL / 8, 256, 0, stream>>>(u, v, divh);
  dyn_core_kernel<<<NCOL / 8, 256, 0, stream>>>(u, v, T, ps, divh, dpsdx, dpsdy, Rp, out);
  zero4_kernel<<<(N4D / 4 + 255) / 256, 256, 0, stream>>>(
      (float4*)(out + (size_t)3 * N4D), N4D / 4);
  polar_filter4d_kernel<<<3 * YDIM * 4, 256, 0, stream>>>(out, htab);
  polar_filter_ps_kernel<<<BDIM * YDIM, 256, 0, stream>>>(out + (size_t)4 * N4D, htab);
}